// ContrastiveCenterLoss_new_33174327394761
// MI455X (gfx1250) — compile-verified
//
#include <hip/hip_runtime.h>
#include <hip/hip_bf16.h>

typedef __attribute__((ext_vector_type(2))) float v2f;
typedef __attribute__((ext_vector_type(8))) float v8f;

#define CCL_EPS 1e-8f

// ---------------------------------------------------------------------------
// Kernel 1: normalize class centers. One wave32 per center (D=128 -> 4 f32
// per lane as float4). cn[c] = fc[c] / max(||fc[c]||, eps)
// ---------------------------------------------------------------------------
__global__ void __launch_bounds__(256)
ccl_normalize_centers(const float* __restrict__ fc, float* __restrict__ cn,
                      int C, int D) {
    const int lane = threadIdx.x & 31;
    const int center = (blockIdx.x * blockDim.x + threadIdx.x) >> 5;
    if (center >= C) return;

    const float4* src = (const float4*)(fc + (size_t)center * D);
    float4* dst = (float4*)(cn + (size_t)center * D);
    const int nq = D >> 2;

    float ss = 0.f;
    for (int q = lane; q < nq; q += 32) {
        float4 v = src[q];
        ss += v.x * v.x + v.y * v.y + v.z * v.z + v.w * v.w;
    }
    // wave32 butterfly reduce
    for (int off = 16; off >= 1; off >>= 1) ss += __shfl_xor(ss, off, 32);
    const float rinv = 1.0f / fmaxf(sqrtf(ss), CCL_EPS);
    for (int q = lane; q < nq; q += 32) {
        float4 v = src[q];
        v.x *= rinv; v.y *= rinv; v.z *= rinv; v.w *= rinv;
        dst[q] = v;
    }
}

// ---------------------------------------------------------------------------
// Kernel 2: s[d] = sum_c cn[c][d]  (C*D = 128K f32, L2-resident, trivial)
// ---------------------------------------------------------------------------
__global__ void ccl_center_sum(const float* __restrict__ cn,
                               float* __restrict__ s, int C, int D) {
    const int d = blockIdx.x * blockDim.x + threadIdx.x;
    if (d >= D) return;
    float acc = 0.f;
    for (int c = 0; c < C; ++c) acc += cn[(size_t)c * D + d];
    s[d] = acc;
}

// ---------------------------------------------------------------------------
// Kernel 3: main WMMA kernel. One wave32 per 16-sample tile; 8 waves / block.
//   accA diag = hidden[b] . cn[y[b]]      (intra, pre-normalization)
//   accS diag = hidden[b] . s             (row-sum of cosine matrix, pre-norm)
//   accN diag = ||hidden[b]||^2
// using V_WMMA_F32_16X16X4_F32 stepped over K=128 (D).
// A 16x4 f32 fragment: lanes 0-15 hold {K=k0,k0+1} of row (lane&15);
// lanes 16-31 hold {K=k0+2,k0+3}. The 4x16 B fragment has the symmetric
// layout with N=(lane&15), so the same index math builds all three operands
// (and the A fragment itself is a valid B fragment -> accN = A*A^T).
// ---------------------------------------------------------------------------
__global__ void __launch_bounds__(256)
ccl_main(const float* __restrict__ hidden, const float* __restrict__ cn,
         const float* __restrict__ s, const long long* __restrict__ y,
         float* __restrict__ partials, int D, float invCm1) {
    __shared__ float wsum[8];

    const int tid  = threadIdx.x;
    const int lane = tid & 31;
    const int wv   = tid >> 5;
    const int tileBase = (blockIdx.x * 8 + wv) * 16;

    const int m  = lane & 15;          // row (A) == column (B) handled by lane
    const int kh = (lane >> 4) << 1;   // 0 for lanes 0-15, 2 for lanes 16-31

    const float* arow = hidden + (size_t)(tileBase + m) * D + kh;
    const int    cls  = (int)y[tileBase + m];
    const float* brow = cn + (size_t)cls * D + kh;
    const float* srow = s + kh;

    v8f accA = {0.f, 0.f, 0.f, 0.f, 0.f, 0.f, 0.f, 0.f};
    v8f accS = {0.f, 0.f, 0.f, 0.f, 0.f, 0.f, 0.f, 0.f};
    v8f accN = {0.f, 0.f, 0.f, 0.f, 0.f, 0.f, 0.f, 0.f};

#pragma unroll
    for (int k0 = 0; k0 < 128; k0 += 4) {   // D == 128
        v2f a  = *(const v2f*)(arow + k0);  // 8B aligned (k0+kh even)
        v2f b  = *(const v2f*)(brow + k0);
        v2f sv = *(const v2f*)(srow + k0);
        // args: (neg_a, A, neg_b, B, c_mod, C, reuse_a, reuse_b)
        accA = __builtin_amdgcn_wmma_f32_16x16x4_f32(false, a, false, b,
                                                     (short)0, accA, false, false);
        accS = __builtin_amdgcn_wmma_f32_16x16x4_f32(false, a, false, sv,
                                                     (short)0, accS, false, false);
        accN = __builtin_amdgcn_wmma_f32_16x16x4_f32(false, a, false, a,
                                                     (short)0, accN, false, false);
    }

    // Diagonal extraction: diag i -> (lane i, vgpr i) for i<8;
    //                      (lane i+16, vgpr i-8) for i>=8.
    const bool act = (lane < 8) || (lane >= 24);
    const int  idx = (lane < 8) ? lane : (lane - 24);
    float l = 0.f;
    if (act) {
        float dA = accA[idx];
        float dS = accS[idx];
        float dN = accN[idx];
        float inv   = 1.0f / fmaxf(sqrtf(dN), CCL_EPS);
        float intra = dA * inv;
        float tot   = dS * inv;
        l = 1.0f - intra + (tot - intra) * invCm1;
    }
    // wave sum (inactive lanes contribute 0)
    for (int off = 16; off >= 1; off >>= 1) l += __shfl_xor(l, off, 32);
    if (lane == 0) wsum[wv] = l;
    __syncthreads();
    if (tid == 0) {
        float t = 0.f;
#pragma unroll
        for (int i = 0; i < 8; ++i) t += wsum[i];
        partials[blockIdx.x] = t;   // deterministic: fixed summation order
    }
}

// ---------------------------------------------------------------------------
// Kernel 4: final deterministic tree reduction of per-block partials.
// ---------------------------------------------------------------------------
__global__ void __launch_bounds__(256)
ccl_reduce(const float* __restrict__ partials, float* __restrict__ out,
           int n, float invB) {
    __shared__ float sm[256];
    const int t = threadIdx.x;
    float a = 0.f;
    for (int i = t; i < n; i += 256) a += partials[i];
    sm[t] = a;
    __syncthreads();
    for (int st = 128; st >= 1; st >>= 1) {
        if (t < st) sm[t] += sm[t + st];
        __syncthreads();
    }
    if (t == 0) out[0] = sm[0] * invB;
}

// ---------------------------------------------------------------------------
extern "C" void kernel_launch(void* const* d_in, const int* in_sizes, int n_in,
                              void* d_out, int out_size, void* d_ws, size_t ws_size,
                              hipStream_t stream) {
    const float*     hidden = (const float*)d_in[0];
    const float*     fc     = (const float*)d_in[1];
    const long long* y      = (const long long*)d_in[2];
    float*           out    = (float*)d_out;

    const int B = in_sizes[2];            // 32768
    const int D = in_sizes[0] / B;        // 128
    const int C = in_sizes[1] / D;        // 1000

    // workspace layout: [cn: C*D f32][s: D f32][partials: NB f32]
    float* cn       = (float*)d_ws;
    float* s        = cn + (size_t)C * D;
    float* partials = s + D;

    const int NB = B / 128;               // 128 samples (8 wave-tiles) / block

    ccl_normalize_centers<<<(C + 7) / 8, 256, 0, stream>>>(fc, cn, C, D);
    ccl_center_sum<<<(D + 127) / 128, 128, 0, stream>>>(cn, s, C, D);
    ccl_main<<<NB, 256, 0, stream>>>(hidden, cn, s, y, partials, D,
                                     1.0f / (float)(C - 1));
    ccl_reduce<<<1, 256, 0, stream>>>(partials, out, NB, 1.0f / (float)B);
}